// MessagePassing_1872605741887
// MI455X (gfx1250) — compile-verified
//
#include <hip/hip_runtime.h>
#include <hip/hip_bf16.h>

// Problem constants (from the reference)
#define BB 8
#define NN 10000
#define FN 256      // F_NODE
#define FE 128      // F_EDGE
#define NE 160000   // N_EDGE

constexpr int M_TOTAL = BB * NN;   // 80000 rows
constexpr int ROWS_PER_BLOCK = 32; // LDS-staged row tile
constexpr int TM = 2;              // 16-row tiles per wave
constexpr int LDS_H_STRIDE = 260;  // 256 + 4 pad -> stride mod 64 banks = 4 (conflict-free)
constexpr int LDS_E_STRIDE = 132;  // 128 + 4 pad

typedef float v2f __attribute__((ext_vector_type(2)));
typedef float v8f __attribute__((ext_vector_type(8)));

// ---------------------------------------------------------------------------
// Kernel 1: scatter-add edge features by source node + degree count.
// One 128-thread block per (b, e): he[b, src, f] += edge_fea[b, e, f].
// edge_fea is streamed once -> non-temporal loads keep L2 free for the 41 MB
// `he` array the atomics hammer (fits in the 192 MB L2).
// ---------------------------------------------------------------------------
__global__ __launch_bounds__(128) void scatter_edges(
    const int*   __restrict__ edge_src,   // (B*NE)
    const float* __restrict__ edge_fea,   // (B*NE, FE)
    float*       __restrict__ he,         // (B*NN, FE)
    float*       __restrict__ deg)        // (B*NN)
{
    const long long be = blockIdx.x;             // 0 .. B*NE-1
    const int b = (int)(be / NE);
    const int s = edge_src[be];                  // source node
    const int f = threadIdx.x;                   // 0..127
    const float v = __builtin_nontemporal_load(&edge_fea[(size_t)be * FE + f]);
    unsafeAtomicAdd(&he[((size_t)b * NN + s) * FE + f], v);
    if (f == 0) unsafeAtomicAdd(&deg[(size_t)b * NN + s], 1.0f);
}

// ---------------------------------------------------------------------------
// Kernel 2: fused fp32 WMMA GEMM with LDS-staged A operands.
// out[m, j] = H@Wself + deg[m]*(H@Wnei_top) + he@Wnei_bot + bias[j]
// Block: 256 threads = 8 waves; covers 32 rows x 128 cols.
//   - H/he row tile staged once into LDS, shared by all 8 waves (8 col tiles).
//   - Each wave: TM=2 16x16 output tiles, 3 accumulator streams (48 VGPRs).
// v_wmma_f32_16x16x4_f32: A 16x4 (v2f), B 4x16 (v2f), C/D 16x16 (v8f).
// ---------------------------------------------------------------------------
__global__ __launch_bounds__(256) void gemm_wmma_f32(
    const float* __restrict__ H,      // (M_TOTAL, FN)
    const float* __restrict__ he,     // (M_TOTAL, FE)
    const float* __restrict__ deg,    // (M_TOTAL)
    const float* __restrict__ Wself,  // (FN, FN) row-major [k][j]
    const float* __restrict__ Wnei,   // (FN+FE, FN) row-major [k][j]
    const float* __restrict__ bias,   // (FN)
    float*       __restrict__ out)    // (M_TOTAL, FN)
{
    __shared__ float ldsH[ROWS_PER_BLOCK * LDS_H_STRIDE]; // 33,280 B
    __shared__ float ldsE[ROWS_PER_BLOCK * LDS_E_STRIDE]; // 16,896 B

    const int tid  = threadIdx.x;
    const int lane = tid & 31;
    const int w    = tid >> 5;                         // wave 0..7 -> col tile
    const int rowBase = blockIdx.x * ROWS_PER_BLOCK;
    const int colBase = blockIdx.y * 128 + w * 16;

    // ---- Cooperative stage: H tile (32 x 256) and he tile (32 x 128) ----
    {
        const float4* H4 = (const float4*)(H + (size_t)rowBase * FN);
#pragma unroll
        for (int it = 0; it < 8; ++it) {               // 2048 float4 chunks
            const int idx = it * 256 + tid;
            const int row = idx >> 6, c4 = idx & 63;
            const float4 v = H4[row * 64 + c4];
            *(float4*)&ldsH[row * LDS_H_STRIDE + c4 * 4] = v;
        }
        const float4* E4 = (const float4*)(he + (size_t)rowBase * FE);
#pragma unroll
        for (int it = 0; it < 4; ++it) {               // 1024 float4 chunks
            const int idx = it * 256 + tid;
            const int row = idx >> 5, c4 = idx & 31;
            const float4 v = E4[row * 32 + c4];
            *(float4*)&ldsE[row * LDS_E_STRIDE + c4 * 4] = v;
        }
    }
    __syncthreads();

    const int ml = lane & 15;    // A: row-in-tile; B/C: col-in-tile
    const int kh = lane >> 4;    // k-half select (K pairs {0,1} vs {2,3})

    v8f accS[TM], accN[TM], accE[TM];
#pragma unroll
    for (int t = 0; t < TM; ++t) {
        accS[t] = (v8f){0.f,0.f,0.f,0.f,0.f,0.f,0.f,0.f};
        accN[t] = (v8f){0.f,0.f,0.f,0.f,0.f,0.f,0.f,0.f};
        accE[t] = (v8f){0.f,0.f,0.f,0.f,0.f,0.f,0.f,0.f};
    }

    // ---- Streams 1 & 2: K = 0..255 over H vs Wself and Wnei(top) ----
    for (int kk = 0; kk < FN; kk += 4) {
        const int k0 = kk + 2 * kh;    // this lane-half supplies K = k0, k0+1
        v2f bS, bN;
        bS.x = Wself[(size_t)(k0    ) * FN + colBase + ml];
        bS.y = Wself[(size_t)(k0 + 1) * FN + colBase + ml];
        bN.x = Wnei [(size_t)(k0    ) * FN + colBase + ml];
        bN.y = Wnei [(size_t)(k0 + 1) * FN + colBase + ml];
#pragma unroll
        for (int t = 0; t < TM; ++t) {
            const v2f a = *(const v2f*)&ldsH[(t * 16 + ml) * LDS_H_STRIDE + k0];
            accS[t] = __builtin_amdgcn_wmma_f32_16x16x4_f32(
                false, a, false, bS, (short)0, accS[t], false, false);
            accN[t] = __builtin_amdgcn_wmma_f32_16x16x4_f32(
                false, a, false, bN, (short)0, accN[t], false, false);
        }
    }

    // ---- Stream 3: K = 0..127 over he vs Wnei rows 256..383 ----
    for (int kk = 0; kk < FE; kk += 4) {
        const int k0 = kk + 2 * kh;
        v2f bN;
        bN.x = Wnei[(size_t)(FN + k0    ) * FN + colBase + ml];
        bN.y = Wnei[(size_t)(FN + k0 + 1) * FN + colBase + ml];
#pragma unroll
        for (int t = 0; t < TM; ++t) {
            const v2f a = *(const v2f*)&ldsE[(t * 16 + ml) * LDS_E_STRIDE + k0];
            accE[t] = __builtin_amdgcn_wmma_f32_16x16x4_f32(
                false, a, false, bN, (short)0, accE[t], false, false);
        }
    }

    // ---- Epilogue: out = accS + deg*accN + accE + bias (streaming stores) ----
    // C/D layout: element i of lane L -> row = i + (L>=16 ? 8 : 0), col = L%16.
    const float bj = bias[colBase + ml];
#pragma unroll
    for (int t = 0; t < TM; ++t) {
        const int rBase = rowBase + t * 16 + kh * 8;
#pragma unroll
        for (int i = 0; i < 8; ++i) {
            const int r = rBase + i;
            const float v = accS[t][i] + deg[r] * accN[t][i] + accE[t][i] + bj;
            __builtin_nontemporal_store(v, &out[(size_t)r * FN + colBase + ml]);
        }
    }
}

// ---------------------------------------------------------------------------
extern "C" void kernel_launch(void* const* d_in, const int* in_sizes, int n_in,
                              void* d_out, int out_size, void* d_ws, size_t ws_size,
                              hipStream_t stream) {
    const float* H        = (const float*)d_in[0];   // (B, N, FN)
    const int*   edge_src = (const int*)  d_in[1];   // (B, NE)
    const float* edge_fea = (const float*)d_in[2];   // (B, NE, FE)
    const float* Wself    = (const float*)d_in[3];   // (FN, FN)
    const float* Wnei     = (const float*)d_in[4];   // (FN+FE, FN)
    const float* bias     = (const float*)d_in[5];   // (FN)
    float* out = (float*)d_out;

    float* he  = (float*)d_ws;                       // (B*NN, FE)
    float* deg = he + (size_t)BB * NN * FE;          // (B*NN)
    const size_t zbytes = ((size_t)BB * NN * FE + (size_t)BB * NN) * sizeof(float);

    hipMemsetAsync(d_ws, 0, zbytes, stream);

    scatter_edges<<<BB * NE, 128, 0, stream>>>(edge_src, edge_fea, he, deg);

    dim3 grid(M_TOTAL / ROWS_PER_BLOCK, 2);          // (2500, 2)
    gemm_wmma_f32<<<grid, 256, 0, stream>>>(H, he, deg, Wself, Wnei, bias, out);
}